// RGIN_87677462381091
// MI455X (gfx1250) — compile-verified
//
#include <hip/hip_runtime.h>

#define NNODES 50000
#define NEDGES 1600000
#define NREL   8
#define DIM    64
#define LDSS   68   // padded per-wave tile row stride (floats)
#define WSTR   68   // padded weight-stage row stride (floats)
#define WAVES  5    // waves per block in node kernel (3125 tiles = 625 * 5)

typedef float v2f __attribute__((ext_vector_type(2)));
typedef float v8f __attribute__((ext_vector_type(8)));

// V_WMMA_F32_16X16X4_F32 : A 16x4 f32 (2 VGPR), B 4x16 f32 (2 VGPR), C/D 16x16 f32 (8 VGPR)
__device__ __forceinline__ v8f wmma4(v2f a, v2f b, v8f c) {
  return __builtin_amdgcn_wmma_f32_16x16x4_f32(
      /*neg_a=*/false, a, /*neg_b=*/false, b,
      /*c_mod=*/(short)0, c, /*reuse_a=*/false, /*reuse_b=*/false);
}

// Async Global -> LDS copy of one 16-byte chunk (ASYNCcnt-tracked DMA path).
__device__ __forceinline__ void async_copy16(const float* g, float* lds) {
  asm volatile("global_load_async_to_lds_b128 %0, %1, off"
               :: "v"((unsigned)(unsigned long long)lds), "v"(g)
               : "memory");
}
__device__ __forceinline__ void wait_async() {
  asm volatile("s_wait_asynccnt 0x0" ::: "memory");
}

// Stage a 64x64 f32 weight matrix into LDS with padded row stride WSTR.
// 1024 x 16B chunks spread over the 160 threads of the block.
__device__ __forceinline__ void stage_weight(const float* __restrict__ g,
                                             float* __restrict__ l, int tid) {
#pragma unroll 1
  for (int i = tid; i < 1024; i += 160) {
    const int row = i >> 4;          // 16 chunks per 64-float row
    const int c4  = (i & 15) << 2;   // float offset within row
    async_copy16(g + (i << 2), l + row * WSTR + c4);
  }
}

// ---------------------------------------------------------------- zero ag ---
__global__ void __launch_bounds__(256) rgin_zero(float4* __restrict__ p, int n4) {
  int i = blockIdx.x * blockDim.x + threadIdx.x;
  const int stride = gridDim.x * blockDim.x;
  const float4 z = make_float4(0.f, 0.f, 0.f, 0.f);
  for (; i < n4; i += stride) p[i] = z;
}

// ------------------------------------------------------------ edge scatter ---
// ag[(type*N + dst)*D + :] += x[src*D + :]   (16 threads per edge, float4 each)
__global__ void __launch_bounds__(256) rgin_scatter(
    const float* __restrict__ xin,
    const int* __restrict__ esrc, const int* __restrict__ edst,
    const int* __restrict__ etyp,
    float* __restrict__ ag, int E)
{
  const int t = blockIdx.x * 256 + threadIdx.x;
  const int e = t >> 4;
  const int q = t & 15;
  if (e >= E) return;
  const int src = esrc[e];
  const int dst = edst[e];
  const int r   = etyp[e];
  const float4 v = ((const float4*)(xin + (size_t)src * DIM))[q];
  float* base = ag + ((size_t)r * NNODES + dst) * DIM + (q << 2);
  unsafeAtomicAdd(base + 0, v.x);
  unsafeAtomicAdd(base + 1, v.y);
  unsafeAtomicAdd(base + 2, v.z);
  unsafeAtomicAdd(base + 3, v.w);
}

// --------------------------------------------------- fused per-node layer ----
// 5 waves / block; each wave owns one 16-node tile end-to-end:
//   msgs = sum_r ag[r] @ W[r]        (W[r] async-staged to LDS, double-buffered)
//   h    = (1+eps)*x + msgs          (-> per-wave LDS tile, C->A re-layout)
//   u    = relu(h @ mw1 + mb1)       (mw1 staged to LDS; tile aliased in place)
//   out  = u @ mw2 + mb2 + bias      (mw2 staged to LDS)
__global__ void __launch_bounds__(160) rgin_node(
    const float* __restrict__ ag, const float* __restrict__ x,
    const float* __restrict__ W,  const float* __restrict__ epsp,
    const float* __restrict__ mw1, const float* __restrict__ mb1,
    const float* __restrict__ mw2, const float* __restrict__ mb2,
    const float* __restrict__ bias, float* __restrict__ out)
{
  __shared__ float wst[2][DIM * WSTR];      // double-buffered weight stage (2x17408B)
  __shared__ float tiles[WAVES][16 * LDSS]; // per-wave h/u tile (aliased, 5x4352B)

  const int tid  = threadIdx.x;
  const int wave = tid >> 5;
  const int lane = tid & 31;
  const int hi   = lane >> 4;
  const int lo   = lane & 15;
  const int krow = hi << 1;                 // K offset within the 4-wide K group
  const int n0   = (blockIdx.x * WAVES + wave) << 4;
  float* myt = &tiles[wave][0];

  v8f acc0[4], acc1[4], acc2[4];
#pragma unroll
  for (int ct = 0; ct < 4; ++ct)
#pragma unroll
    for (int i = 0; i < 8; ++i) { acc0[ct][i] = 0.f; acc1[ct][i] = 0.f; acc2[ct][i] = 0.f; }

  // Prologue: stage W[0]
  stage_weight(W, &wst[0][0], tid);
  wait_async();
  __syncthreads();

  // ---- GEMM0: msgs = sum_r ag[r] @ W[r] -------------------------------------
#pragma unroll 1
  for (int r = 0; r < NREL; ++r) {
    const float* wl = &wst[r & 1][0];
    if (r + 1 < NREL)  // prefetch next relation's weights while computing
      stage_weight(W + ((r + 1) << 12), &wst[(r + 1) & 1][0], tid);
    const float* agrow = ag + ((size_t)r * NNODES + n0 + lo) * DIM + krow;
#pragma unroll 1
    for (int k = 0; k < DIM; k += 4) {
      const v2f a = *(const v2f*)(agrow + k);
#pragma unroll
      for (int ct = 0; ct < 4; ++ct) {
        const int col = (ct << 4) + lo;
        v2f b;
        b.x = wl[(k + krow) * WSTR + col];
        b.y = wl[(k + krow + 1) * WSTR + col];
        acc0[ct] = wmma4(a, b, acc0[ct]);
      }
    }
    wait_async();
    __syncthreads();
  }

  // Stage MLP weights into the (now free) two buffers; overlap with epilogue.
  stage_weight(mw1, &wst[0][0], tid);
  stage_weight(mw2, &wst[1][0], tid);

  // ---- h = (1+eps)*x + msgs  -> per-wave LDS tile ---------------------------
  const float se = 1.0f + epsp[0];
#pragma unroll
  for (int ct = 0; ct < 4; ++ct) {
    const int col = (ct << 4) + lo;
#pragma unroll
    for (int v = 0; v < 8; ++v) {
      const int row = v + (hi << 3);
      myt[row * LDSS + col] = acc0[ct][v] + se * x[(size_t)(n0 + row) * DIM + col];
    }
  }
  wait_async();
  __syncthreads();

  // ---- GEMM1: u = relu(h @ mw1 + mb1) ---------------------------------------
#pragma unroll 1
  for (int k = 0; k < DIM; k += 4) {
    v2f a;
    a.x = myt[lo * LDSS + k + krow];
    a.y = myt[lo * LDSS + k + krow + 1];
#pragma unroll
    for (int ct = 0; ct < 4; ++ct) {
      const int col = (ct << 4) + lo;
      v2f b;
      b.x = wst[0][(k + krow) * WSTR + col];
      b.y = wst[0][(k + krow + 1) * WSTR + col];
      acc1[ct] = wmma4(a, b, acc1[ct]);
    }
  }
  // relu + bias, write back into the same per-wave tile (in-wave LDS ordering)
#pragma unroll
  for (int ct = 0; ct < 4; ++ct) {
    const int col = (ct << 4) + lo;
    const float b1 = mb1[col];
#pragma unroll
    for (int v = 0; v < 8; ++v) {
      const int row = v + (hi << 3);
      myt[row * LDSS + col] = fmaxf(acc1[ct][v] + b1, 0.0f);
    }
  }

  // ---- GEMM2: out = u @ mw2 + mb2 + bias ------------------------------------
#pragma unroll 1
  for (int k = 0; k < DIM; k += 4) {
    v2f a;
    a.x = myt[lo * LDSS + k + krow];
    a.y = myt[lo * LDSS + k + krow + 1];
#pragma unroll
    for (int ct = 0; ct < 4; ++ct) {
      const int col = (ct << 4) + lo;
      v2f b;
      b.x = wst[1][(k + krow) * WSTR + col];
      b.y = wst[1][(k + krow + 1) * WSTR + col];
      acc2[ct] = wmma4(a, b, acc2[ct]);
    }
  }
#pragma unroll
  for (int ct = 0; ct < 4; ++ct) {
    const int col = (ct << 4) + lo;
    const float bb = mb2[col] + bias[col];
#pragma unroll
    for (int v = 0; v < 8; ++v) {
      const int row = v + (hi << 3);
      out[(size_t)(n0 + row) * DIM + col] = acc2[ct][v] + bb;
    }
  }
}

// ---------------------------------------------------------------------------
extern "C" void kernel_launch(void* const* d_in, const int* in_sizes, int n_in,
                              void* d_out, int out_size, void* d_ws, size_t ws_size,
                              hipStream_t stream)
{
  const float* x     = (const float*)d_in[0];
  const int*   esrc  = (const int*)d_in[1];
  const int*   edst  = (const int*)d_in[2];
  const int*   etyp  = (const int*)d_in[3];
  const float* W1    = (const float*)d_in[4];
  const float* bias1 = (const float*)d_in[5];
  const float* eps1  = (const float*)d_in[6];
  const float* m1w1  = (const float*)d_in[7];
  const float* m1b1  = (const float*)d_in[8];
  const float* m1w2  = (const float*)d_in[9];
  const float* m1b2  = (const float*)d_in[10];
  const float* W2    = (const float*)d_in[11];
  const float* bias2 = (const float*)d_in[12];
  const float* eps2  = (const float*)d_in[13];
  const float* m2w1  = (const float*)d_in[14];
  const float* m2b1  = (const float*)d_in[15];
  const float* m2w2  = (const float*)d_in[16];
  const float* m2b2  = (const float*)d_in[17];

  float* ag  = (float*)d_ws;                              // R*N*D floats (102.4 MB, L2-resident)
  float* h1  = ag + (size_t)NREL * NNODES * DIM;          // N*D floats
  float* out = (float*)d_out;

  const int agN = NREL * NNODES * DIM;                    // 25,600,000
  const dim3 zgrid(agN / 4 / 256);                        // 25,000 blocks
  const dim3 sgrid((NEDGES * 16) / 256);                  // 100,000 blocks
  const dim3 ngrid(NNODES / 16 / WAVES);                  // 625 blocks x 5 waves

  // ---- layer 1 ----
  rgin_zero   <<<zgrid, 256, 0, stream>>>((float4*)ag, agN / 4);
  rgin_scatter<<<sgrid, 256, 0, stream>>>(x, esrc, edst, etyp, ag, NEDGES);
  rgin_node   <<<ngrid, 160, 0, stream>>>(ag, x, W1, eps1, m1w1, m1b1, m1w2, m1b2, bias1, h1);

  // ---- layer 2 ----
  rgin_zero   <<<zgrid, 256, 0, stream>>>((float4*)ag, agN / 4);
  rgin_scatter<<<sgrid, 256, 0, stream>>>(h1, esrc, edst, etyp, ag, NEDGES);
  rgin_node   <<<ngrid, 160, 0, stream>>>(ag, h1, W2, eps2, m2w1, m2b1, m2w2, m2b2, bias2, out);
}